// Model_40690520162338
// MI455X (gfx1250) — compile-verified
//
#include <hip/hip_runtime.h>

typedef __attribute__((ext_vector_type(2))) float v2f;
typedef __attribute__((ext_vector_type(4))) float v4f;
typedef __attribute__((ext_vector_type(8))) float v8f;

#define B_    128
#define S_    672
#define P_    168
#define C_    500
#define KB    32
#define NK    (S_ / KB)     // 21
#define NTILE 64
#define NT_N  3             // ceil(168/64)
#define LDA   36            // padded LDS row stride (dwords): conflict-free + 16B aligned
#define LDB   36

// Workgroup barrier that makes prior LDS stores visible WITHOUT flushing
// LOADcnt (unlike __syncthreads, which emits s_wait_loadcnt_dscnt 0 and kills
// the global-load / WMMA overlap). Next-tile global loads stay in flight
// across the barrier; their wait lands at the LDS store via register deps.
__device__ __forceinline__ void wg_barrier_lds_only() {
    asm volatile(
        "s_wait_dscnt 0x0\n\t"
        "s_barrier_signal -1\n\t"
        "s_barrier_wait -1"
        ::: "memory");
}

__launch_bounds__(256, 2)
__global__ void chan_gemm_f32_wmma(const float* __restrict__ x,
                                   const float* __restrict__ W,
                                   const float* __restrict__ bias,
                                   float* __restrict__ out) {
    __shared__ float lA[2][B_ * LDA];      // 2 x 128 x 36 x 4B = 36.9 KB
    __shared__ float lB[2][NTILE * LDB];   // 2 x  64 x 36 x 4B = 18.4 KB
    __shared__ float lLast[B_];

    const int t    = threadIdx.x;
    const int c    = blockIdx.x % C_;            // channel-fastest => L2 line reuse on x
    const int n0   = (blockIdx.x / C_) * NTILE;  // P-tile base

    const int lane = t & 31;
    const int wave = t >> 5;
    const int mw   = wave >> 1;   // 0..3 : 32-row band of B
    const int nw   = wave & 1;    // 0..1 : 32-col band of P-tile

    // ---- seq_last staging: last[b] = x[b, S-1, c] ----
    const int   brow  = t >> 1;                               // 0..127
    const float lastb = x[brow * S_ * C_ + (S_ - 1) * C_ + c];
    if ((t & 1) == 0) lLast[brow] = lastb;

    // A staging: thread t -> x[brow, k0 + sHalf + j, c], j=0..15 (strided gather)
    const int sHalf = (t & 1) * 16;
    // B staging: thread t -> W[c, n0 + pRow, k0 + sQuad .. +7] (contiguous float4s)
    const int pRow  = t >> 2;                                 // 0..63
    const int pEff  = (n0 + pRow < P_) ? (n0 + pRow) : (P_ - 1);
    const int sQuad = (t & 3) * 8;
    const float* Wrow = W + (size_t)c * P_ * S_ + (size_t)pEff * S_;

    float aReg[16];
    v4f   bReg[2];

    auto loadTile = [&](int kt) {
        const int k0 = kt * KB;
        const float* xp = x + brow * S_ * C_ + (k0 + sHalf) * C_ + c;
        #pragma unroll
        for (int j = 0; j < 16; ++j) aReg[j] = xp[j * C_];
        const float* wp = Wrow + k0 + sQuad;
        bReg[0] = *(const v4f*)(wp);
        bReg[1] = *(const v4f*)(wp + 4);
    };
    auto storeTile = [&](int buf) {
        float* ar = &lA[buf][brow * LDA + sHalf];
        #pragma unroll
        for (int j = 0; j < 16; ++j) ar[j] = aReg[j] - lastb;  // fold (x - seq_last)
        float* br = &lB[buf][pRow * LDB + sQuad];
        *(v4f*)(br)     = bReg[0];
        *(v4f*)(br + 4) = bReg[1];
    };

    v8f acc00 = {}, acc01 = {}, acc10 = {}, acc11 = {};

    // WMMA f32 16x16x4 fragment mapping (per ISA 7.12.2):
    //   A: lane m = lane&15, VGPR0/1 = K {kb, kb+1}, kb = (lane>=16)?2:0
    //   B (K x N): mirrored: lane n = lane&15, K {kb, kb+1}
    const int mfrag = lane & 15;
    const int kb    = (lane >> 4) * 2;

    loadTile(0);
    storeTile(0);
    int buf = 0;

    for (int kt = 0; kt < NK; ++kt) {
        if (kt + 1 < NK) loadTile(kt + 1);             // issue next-tile globals
        if (kt + 2 < NK) __builtin_prefetch(Wrow + (kt + 2) * KB + sQuad, 0, 1);
        wg_barrier_lds_only();                         // DScnt-only barrier:
                                                       // globals stay in flight
        const float* Abase = &lA[buf][(mw * 32) * LDA + kb];
        const float* Bbase = &lB[buf][(nw * 32) * LDB + kb];
        #pragma unroll
        for (int kk = 0; kk < KB; kk += 4) {
            v2f a0 = *(const v2f*)(Abase + (mfrag     ) * LDA + kk);
            v2f a1 = *(const v2f*)(Abase + (mfrag + 16) * LDA + kk);
            v2f b0 = *(const v2f*)(Bbase + (mfrag     ) * LDB + kk);
            v2f b1 = *(const v2f*)(Bbase + (mfrag + 16) * LDB + kk);
            acc00 = __builtin_amdgcn_wmma_f32_16x16x4_f32(false, a0, false, b0, (short)0, acc00, false, false);
            acc01 = __builtin_amdgcn_wmma_f32_16x16x4_f32(false, a0, false, b1, (short)0, acc01, false, false);
            acc10 = __builtin_amdgcn_wmma_f32_16x16x4_f32(false, a1, false, b0, (short)0, acc10, false, false);
            acc11 = __builtin_amdgcn_wmma_f32_16x16x4_f32(false, a1, false, b1, (short)0, acc11, false, false);
        }
        if (kt + 1 < NK) storeTile(buf ^ 1);           // s_wait_loadcnt lands here,
                                                       // after the 32 WMMAs
        buf ^= 1;
    }

    // ---- epilogue: out[b,p,c] = acc + bias[c,p] + last[b] ----
    const float* biasC = bias + (size_t)c * P_;
    const int mhalf = (lane >> 4) * 8;                 // C/D layout: lanes16-31 hold M=i+8
    #pragma unroll
    for (int sm = 0; sm < 2; ++sm) {
        #pragma unroll
        for (int sn = 0; sn < 2; ++sn) {
            const v8f acc  = (sm == 0) ? (sn == 0 ? acc00 : acc01)
                                       : (sn == 0 ? acc10 : acc11);
            const int  nAbs  = n0 + nw * 32 + sn * 16 + (lane & 15);
            const bool valid = (nAbs < P_);
            const float bv   = valid ? biasC[nAbs] : 0.0f;
            #pragma unroll
            for (int i = 0; i < 8; ++i) {
                const int mAbs = mw * 32 + sm * 16 + mhalf + i;
                const float r  = acc[i] + bv + lLast[mAbs];
                if (valid)
                    out[(size_t)mAbs * P_ * C_ + (size_t)nAbs * C_ + c] = r;
            }
        }
    }
}

extern "C" void kernel_launch(void* const* d_in, const int* in_sizes, int n_in,
                              void* d_out, int out_size, void* d_ws, size_t ws_size,
                              hipStream_t stream) {
    (void)in_sizes; (void)n_in; (void)out_size; (void)d_ws; (void)ws_size;
    const float* x  = (const float*)d_in[0];
    const float* W  = (const float*)d_in[1];
    const float* bi = (const float*)d_in[2];
    float* out      = (float*)d_out;

    dim3 grid(C_ * NT_N);   // 1500 workgroups, channel-fastest ordering
    dim3 block(256);        // 8 wave32s
    chan_gemm_f32_wmma<<<grid, block, 0, stream>>>(x, W, bi, out);
}